// KMeansPP_26594437496889
// MI455X (gfx1250) — compile-verified
//
#include <hip/hip_runtime.h>
#include <math.h>

// KMeans++ init on gfx1250 (MI455X).
// One persistent workgroup per batch: data (128MB) is L2-resident, each step
// streams the batch slice (4MB) from L2. Distances via d = |x|^2 - 2 x.c + |c|^2,
// where BOTH dot products come from V_WMMA_F32_16X16X4_F32:
//   accD = wmma(A, Cbroadcast)  -> x.c in every column
//   accG = wmma(A, A)           -> Gram matrix; diagonal = |x|^2
// (for f32 4x16 the B-fragment layout is bit-identical to the A-fragment, so
//  reusing A as B is free — no norm array, no extra loads).

#define B_    32
#define N_    16384
#define D_    64
#define K_    64
#define TPB   512              // 16 wave32s
#define WAVES (TPB / 32)
#define NTILES (N_ / 16)       // 1024 tiles of 16 points

typedef __attribute__((ext_vector_type(2))) float v2f;
typedef __attribute__((ext_vector_type(8))) float v8f;

__launch_bounds__(TPB)
__global__ void kmeanspp_gfx1250_kernel(const float* __restrict__ data,
                                        const int*   __restrict__ first_idx,
                                        float*       __restrict__ out)
{
    __shared__ float minD[N_];      // 64KB running min squared distance
    __shared__ float cen[D_];       // current center
    __shared__ float redV[WAVES];
    __shared__ int   redI[WAVES];
    __shared__ int   sIdx;
    __shared__ float sCnorm;

    const int b    = blockIdx.x;
    const int tid  = threadIdx.x;
    const int lane = tid & 31;
    const int wav  = tid >> 5;

    const float* dbase = data + (size_t)b * N_ * D_;

    // running-min starts at +inf
    for (int p = tid; p < N_; p += TPB) minD[p] = 3.402823466e38f;
    __syncthreads();

    // A-fragment addressing (ISA 16x4 f32 layout):
    //   lanes 0-15: point = frow, dims {4c,4c+1};  lanes 16-31: dims {4c+2,4c+3}
    const int frow = lane & 15;
    const int koff = (lane >> 4) << 1;          // 0 or 2

    // Diagonal owners of the 16x16 C tile: lanes 0-7 -> points 0-7 (acc[lane]),
    // lanes 24-31 -> points 8-15 (acc[lane-24]).
    const bool act  = (lane < 8) || (lane >= 24);
    const int  rsel = (lane < 8) ? lane : (lane - 24);  // accumulator element
    const int  moff = (lane < 8) ? lane : (lane - 16);  // point within tile

    float best  = -3.402823466e38f;
    int   bestI = 0;

    for (int k = 0; k < K_; ++k) {
        int idx;
        if (k == 0) {
            idx = first_idx[b];
        } else {
            // argmax over fresh min_d values tracked by each thread last step
            float bv = best; int bi = bestI;
            #pragma unroll
            for (int off = 16; off > 0; off >>= 1) {
                float ov = __shfl_down(bv, off, 32);
                int   oi = __shfl_down(bi, off, 32);
                if (ov > bv || (ov == bv && oi < bi)) { bv = ov; bi = oi; }
            }
            if (lane == 0) { redV[wav] = bv; redI[wav] = bi; }
            __syncthreads();
            if (tid == 0) {
                float mv = redV[0]; int mi = redI[0];
                for (int w = 1; w < WAVES; ++w)
                    if (redV[w] > mv || (redV[w] == mv && redI[w] < mi)) { mv = redV[w]; mi = redI[w]; }
                sIdx = mi;
            }
            __syncthreads();
            idx = sIdx;
        }

        // stage the chosen center in LDS and emit it to the output
        if (tid < D_) {
            float v = dbase[(size_t)idx * D_ + tid];
            cen[tid] = v;
            out[((size_t)b * K_ + k) * D_ + tid] = v;
        }
        __syncthreads();
        if (tid < 32) {                                  // |c|^2 via wave 0
            float q = cen[tid] * cen[tid] + cen[tid + 32] * cen[tid + 32];
            #pragma unroll
            for (int off = 16; off > 0; off >>= 1) q += __shfl_down(q, off, 32);
            if (tid == 0) sCnorm = q;
        }
        __syncthreads();
        const float cnorm = sCnorm;

        // B-fragments of the broadcast center: every column identical
        v2f kb[16];
        #pragma unroll
        for (int c = 0; c < 16; ++c) {
            kb[c].x = cen[4 * c + koff];
            kb[c].y = cen[4 * c + 1 + koff];
        }

        best = -3.402823466e38f; bestI = 0;

        // stream tiles of 16 points, strided across the 16 waves
        for (int t = wav; t < NTILES; t += WAVES) {
            const int tb = t << 4;
            const float* pA = dbase + (size_t)(tb + frow) * D_ + koff;
            // WGP-scope prefetch of the next tile: pull into near caches so the
            // next iteration's b64 loads hit WGP$ (data itself lives in GL2).
            __builtin_prefetch(pA + (size_t)WAVES * 16 * D_, 0, 3);

            v8f accD = {};          // x . c
            v8f accG = {};          // x . x (diag)
            #pragma unroll
            for (int c = 0; c < 16; ++c) {
                v2f a = *(const v2f*)(pA + 4 * c);
                accD = __builtin_amdgcn_wmma_f32_16x16x4_f32(
                           false, a, false, kb[c], (short)0, accD, false, false);
                accG = __builtin_amdgcn_wmma_f32_16x16x4_f32(
                           false, a, false, a,     (short)0, accG, false, false);
            }

            if (act) {
                // select accumulator element rsel (0..7) via cndmask chain
                float u0 = (rsel & 1) ? accD[1] : accD[0];
                float u1 = (rsel & 1) ? accD[3] : accD[2];
                float u2 = (rsel & 1) ? accD[5] : accD[4];
                float u3 = (rsel & 1) ? accD[7] : accD[6];
                float v0 = (rsel & 2) ? u1 : u0;
                float v1 = (rsel & 2) ? u3 : u2;
                float sD = (rsel & 4) ? v1 : v0;

                float g0 = (rsel & 1) ? accG[1] : accG[0];
                float g1 = (rsel & 1) ? accG[3] : accG[2];
                float g2 = (rsel & 1) ? accG[5] : accG[4];
                float g3 = (rsel & 1) ? accG[7] : accG[6];
                float w0 = (rsel & 2) ? g1 : g0;
                float w1 = (rsel & 2) ? g3 : g2;
                float sG = (rsel & 4) ? w1 : w0;

                const int p = tb + moff;
                float dsq = fmaxf(sG - 2.0f * sD + cnorm, 0.0f);
                float nm  = fminf(minD[p], dsq);
                minD[p] = nm;                        // conflict-free: banks 0..15
                if (nm > best || (nm == best && p < bestI)) { best = nm; bestI = p; }
            }
        }
        // minD[p] is private to its owning thread; barriers above the next
        // center load order everything else.
    }
}

extern "C" void kernel_launch(void* const* d_in, const int* in_sizes, int n_in,
                              void* d_out, int out_size, void* d_ws, size_t ws_size,
                              hipStream_t stream) {
    (void)in_sizes; (void)n_in; (void)out_size; (void)d_ws; (void)ws_size;
    const float* data      = (const float*)d_in[0];
    const int*   first_idx = (const int*)d_in[1];
    float*       out       = (float*)d_out;
    kmeanspp_gfx1250_kernel<<<dim3(B_), dim3(TPB), 0, stream>>>(data, first_idx, out);
}